// FullTensorProduct_44848048505012
// MI455X (gfx1250) — compile-verified
//
#include <hip/hip_runtime.h>

// O(2) FullTensorProduct: input1 (B,448) x input2 (B,160) -> out (B,71680)
// Store-bandwidth-bound (587 MB out). Outer products done with
// V_WMMA_F32_16X16X4_F32: term0 on K=0 (lanes 0-15), term1 on K=2 (lanes 16-31).

typedef __attribute__((ext_vector_type(2))) float v2f;
typedef __attribute__((ext_vector_type(8))) float v8f;

#define RT2  0.70710678118654752440f   // 1/sqrt(2)
#define SQ2  1.41421356237309504880f   // sqrt(2)

__device__ __forceinline__ v8f outer_wmma(float a, float b) {
    v2f A; A.x = a; A.y = 0.0f;       // A[m, K=0 or K=2] per lane-half
    v2f B; B.x = b; B.y = 0.0f;       // B[K=0 or K=2, n] per lane-half
    v8f C = {};
    // 8 args: (neg_a, A, neg_b, B, c_mod, C, reuse_a, reuse_b)
    return __builtin_amdgcn_wmma_f32_16x16x4_f32(
        false, A, false, B, (short)0, C, false, false);
}

// One rank<=2 16x16 outer-product block for one output component.
// half=0 lanes carry term (c0,i0,j0) on K=0; half=1 lanes carry (c1,i1,j1) on K=2.
__device__ __forceinline__ v8f path_k(const float* s1, const float* s2,
                                      int u, int v, int half,
                                      int off1, int d1, int off2, int d2,
                                      float c0, int i0, int j0,
                                      float c1, int i1, int j1) {
    const float c  = half ? c1 : c0;
    const int   ii = half ? i1 : i0;
    const int   jj = half ? j1 : j0;
    const float a = c * s1[off1 + u * d1 + ii];
    const float b =     s2[off2 + v * d2 + jj];
    return outer_wmma(a, b);
}

__device__ __forceinline__ void store_d1(float* __restrict__ outp, int obase,
                                         int mbase, int n, v8f D) {
#pragma unroll
    for (int r = 0; r < 8; ++r)
        outp[obase + (mbase + r) * 32 + n] = D[r];
}

__device__ __forceinline__ void store_d2(float* __restrict__ outp, int obase,
                                         int mbase, int n, v8f Da, v8f Db) {
#pragma unroll
    for (int r = 0; r < 8; ++r) {
        v2f val; val.x = Da[r]; val.y = Db[r];
        *(v2f*)(outp + obase + ((mbase + r) * 32 + n) * 2) = val;
    }
}

#define PATH1(OFF1, D1, OFF2, D2, OB, C0, I0, J0, C1, I1, J1)                \
    store_d1(outp, OB, mbase, n,                                             \
             path_k(s1, s2, u, v, half, OFF1, D1, OFF2, D2,                  \
                    C0, I0, J0, C1, I1, J1));

#define PATH2(OFF1, D1, OFF2, D2, OB,                                        \
              C0a, I0a, J0a, C1a, I1a, J1a,                                  \
              C0b, I0b, J0b, C1b, I1b, J1b)                                  \
    {                                                                        \
        v8f Dk0 = path_k(s1, s2, u, v, half, OFF1, D1, OFF2, D2,             \
                         C0a, I0a, J0a, C1a, I1a, J1a);                      \
        v8f Dk1 = path_k(s1, s2, u, v, half, OFF1, D1, OFF2, D2,             \
                         C0b, I0b, J0b, C1b, I1b, J1b);                      \
        store_d2(outp, OB, mbase, n, Dk0, Dk1);                              \
    }

__global__ __launch_bounds__(256) void o2_tp_wmma_kernel(
    const float* __restrict__ in1, const float* __restrict__ in2,
    float* __restrict__ out) {
    __shared__ float s1[448];
    __shared__ float s2[160];

    const int b   = blockIdx.x;
    const int tid = threadIdx.x;

    for (int t = tid; t < 448; t += 256) s1[t] = in1[b * 448 + t];
    for (int t = tid; t < 160; t += 256) s2[t] = in2[b * 160 + t];
    __syncthreads();

    const int lane  = tid & 31;           // wave32
    const int wave  = tid >> 5;           // 0..7
    const int utile = wave >> 1;          // 0..3  (u tiles of 16 over 64 muls)
    const int vtile = wave & 1;           // 0..1  (v tiles of 16 over 32 muls)
    const int half  = lane >> 4;          // 0: K=0 term, 1: K=2 term
    const int l     = lane & 15;
    const int u     = utile * 16 + l;     // x1 mul index held by this lane (A row)
    const int v     = vtile * 16 + l;     // x2 mul index held by this lane (B col)
    const int n     = vtile * 16 + l;     // D column this lane stores
    const int mbase = utile * 16 + half * 8;  // D row base (M=r or r+8)

    float* outp = out + (size_t)b * 71680;

    // ---- d_out = 1 paths (scalar / pseudoscalar outputs) ----
    PATH1(  0, 1,  0, 1,     0, 1.0f, 0, 0,  0.0f, 0, 0)   // 0x0 -> 0e
    PATH1( 64, 2, 32, 2,  2048,  RT2, 0, 0,   RT2, 1, 1)   // 1x1 -> 0e (cos)
    PATH1(192, 2, 96, 2,  4096,  RT2, 0, 0,   RT2, 1, 1)   // 2x2 -> 0e (cos)
    PATH1( 64, 2, 32, 2,  6144,  RT2, 1, 0,  -RT2, 0, 1)   // 1x1 -> 0o (sin)
    PATH1(192, 2, 96, 2,  8192,  RT2, 1, 0,  -RT2, 0, 1)   // 2x2 -> 0o (sin)

    // ---- d_out = 2 paths, output order sorted by (m_out, -parity) ----
    // m_out = 1
    PATH2(  0, 1, 32, 2, 10240,  SQ2, 0, 0, 0.f, 0, 0,   SQ2, 0, 1, 0.f, 0, 0) // 0x1
    PATH2( 64, 2,  0, 1, 14336,  SQ2, 0, 0, 0.f, 0, 0,   SQ2, 1, 0, 0.f, 0, 0) // 1x0
    PATH2( 64, 2, 96, 2, 18432,  1.f, 0, 0, 1.f, 1, 1,  -1.f, 1, 0, 1.f, 0, 1) // 1x2 diff
    PATH2(192, 2, 32, 2, 22528,  1.f, 0, 0, 1.f, 1, 1,   1.f, 1, 0,-1.f, 0, 1) // 2x1 diff
    PATH2(320, 2, 96, 2, 26624,  1.f, 0, 0, 1.f, 1, 1,   1.f, 1, 0,-1.f, 0, 1) // 3x2 diff
    // m_out = 2
    PATH2(  0, 1, 96, 2, 30720,  SQ2, 0, 0, 0.f, 0, 0,   SQ2, 0, 1, 0.f, 0, 0) // 0x2
    PATH2( 64, 2, 32, 2, 34816,  1.f, 0, 0,-1.f, 1, 1,   1.f, 1, 0, 1.f, 0, 1) // 1x1 sum
    PATH2(192, 2,  0, 1, 38912,  SQ2, 0, 0, 0.f, 0, 0,   SQ2, 1, 0, 0.f, 0, 0) // 2x0
    PATH2(320, 2, 32, 2, 43008,  1.f, 0, 0, 1.f, 1, 1,   1.f, 1, 0,-1.f, 0, 1) // 3x1 diff
    // m_out = 3
    PATH2( 64, 2, 96, 2, 47104,  1.f, 0, 0,-1.f, 1, 1,   1.f, 1, 0, 1.f, 0, 1) // 1x2 sum
    PATH2(192, 2, 32, 2, 51200,  1.f, 0, 0,-1.f, 1, 1,   1.f, 1, 0, 1.f, 0, 1) // 2x1 sum
    PATH2(320, 2,  0, 1, 55296,  SQ2, 0, 0, 0.f, 0, 0,   SQ2, 1, 0, 0.f, 0, 0) // 3x0
    // m_out = 4
    PATH2(192, 2, 96, 2, 59392,  1.f, 0, 0,-1.f, 1, 1,   1.f, 1, 0, 1.f, 0, 1) // 2x2 sum
    PATH2(320, 2, 32, 2, 63488,  1.f, 0, 0,-1.f, 1, 1,   1.f, 1, 0, 1.f, 0, 1) // 3x1 sum
    // m_out = 5
    PATH2(320, 2, 96, 2, 67584,  1.f, 0, 0,-1.f, 1, 1,   1.f, 1, 0, 1.f, 0, 1) // 3x2 sum
}

extern "C" void kernel_launch(void* const* d_in, const int* in_sizes, int n_in,
                              void* d_out, int out_size, void* d_ws, size_t ws_size,
                              hipStream_t stream) {
    (void)n_in; (void)out_size; (void)d_ws; (void)ws_size;
    const float* in1 = (const float*)d_in[0];
    const float* in2 = (const float*)d_in[1];
    float* out = (float*)d_out;
    const int batch = in_sizes[0] / 448;   // 2048
    o2_tp_wmma_kernel<<<batch, 256, 0, stream>>>(in1, in2, out);
}